// Net_19662360281543
// MI455X (gfx1250) — compile-verified
//
#include <hip/hip_runtime.h>

typedef __attribute__((ext_vector_type(2))) float v2f;
typedef __attribute__((ext_vector_type(8))) float v8f;

#define NT 28      // time steps (image columns)
#define NI 28      // input features per step (image rows)
#define NH 32      // hidden width
#define NO 10      // output width (padded to 16 in WMMA N)
#define BT 16      // batch tile per wave (WMMA M)

#define XSTR 449               // LDS stride per t: 28*16 + 1 pad (bank-conflict free)
#define XS_LEN (NT * XSTR)     // 12572 floats
#define SSTR 17                // spike LDS stride per hidden unit (pad)
#define SL_LEN (NH * SSTR)     // 544 floats

__global__ __launch_bounds__(32) void snn_if_mlp(
    const float* __restrict__ x,
    const float* __restrict__ W1, const float* __restrict__ b1,
    const float* __restrict__ W2, const float* __restrict__ b2,
    float* __restrict__ out)
{
  __shared__ float xs[XS_LEN];   // staged image tile, layout [t][i][b]
  __shared__ float sl[SL_LEN];   // layer-1 spikes, layout [o][b]

  const int lane = threadIdx.x;          // 0..31 (wave32)
  const int half = lane >> 4;            // lane half selects K pair (k += 2) / M-high
  const int m    = lane & 15;            // M (A) / N (B,C,D) index
  const size_t tb = (size_t)blockIdx.x * BT;

  // ---- stage 16 fp32 images global -> LDS, transposed to [t][i][b] ----
  const float* src = x + tb * (size_t)(NI * NT);
  for (int idx = lane; idx < BT * NI * NT; idx += 32) {
    int b = idx / (NI * NT);
    int r = idx - b * (NI * NT);
    int i = r / NT;
    int t = r - i * NT;
    xs[t * XSTR + i * 16 + b] = src[idx];
  }

  // ---- preload weights in WMMA B-operand layout ----
  // B[k][n]: vgpr p holds k = 4j + p + 2*half, n = lane&15
  v2f Bw1[2][7];
#pragma unroll
  for (int nt = 0; nt < 2; ++nt) {
    const int o = m + 16 * nt;
#pragma unroll
    for (int j = 0; j < 7; ++j) {
      const int k = 4 * j + 2 * half;
      v2f w; w.x = W1[o * NI + k]; w.y = W1[o * NI + k + 1];
      Bw1[nt][j] = w;
    }
  }
  v2f Bw2[8];
#pragma unroll
  for (int j = 0; j < 8; ++j) {
    v2f w; w.x = 0.0f; w.y = 0.0f;
    if (m < NO) {
      const int k = 4 * j + 2 * half;
      w.x = W2[m * NH + k]; w.y = W2[m * NH + k + 1];
    }
    Bw2[j] = w;
  }
  const float b1v0 = b1[m];
  const float b1v1 = b1[m + 16];
  const float b2v  = (m < NO) ? b2[m] : 0.0f;

  __syncthreads();

  // IF membrane state + firing-rate accumulator live in C-layout registers
  v8f v1a = {}, v1b = {}, vm2 = {}, rate = {};

  for (int t = 0; t < NT; ++t) {
    // ---- layer 1: H1 = X_t(16x28) * W1^T(28x32) + b1  (14 WMMAs) ----
    v8f acc0, acc1;
#pragma unroll
    for (int r = 0; r < 8; ++r) { acc0[r] = b1v0; acc1[r] = b1v1; }
    const float* xcol = &xs[t * XSTR + 32 * half + m];
#pragma unroll
    for (int j = 0; j < 7; ++j) {
      v2f a; a.x = xcol[64 * j]; a.y = xcol[64 * j + 16];  // A[m][4j+2h], A[m][4j+1+2h]
      acc0 = __builtin_amdgcn_wmma_f32_16x16x4_f32(false, a, false, Bw1[0][j],
                                                   (short)0, acc0, false, false);
      acc1 = __builtin_amdgcn_wmma_f32_16x16x4_f32(false, a, false, Bw1[1][j],
                                                   (short)0, acc1, false, false);
    }

    // ---- layer-1 IF neuron: v += h; s = H(v-1); hard reset ----
    float sp0[8], sp1[8];
#pragma unroll
    for (int r = 0; r < 8; ++r) {
      v1a[r] += acc0[r];
      sp0[r] = (v1a[r] >= 1.0f) ? 1.0f : 0.0f;
      if (v1a[r] >= 1.0f) v1a[r] = 0.0f;
      v1b[r] += acc1[r];
      sp1[r] = (v1b[r] >= 1.0f) ? 1.0f : 0.0f;
      if (v1b[r] >= 1.0f) v1b[r] = 0.0f;
    }

    __syncthreads();  // WAR: previous iteration's spike reads done
#pragma unroll
    for (int r = 0; r < 8; ++r) {
      const int b = r + 8 * half;            // C-layout M index
      sl[m * SSTR + b]        = sp0[r];      // o = m
      sl[(m + 16) * SSTR + b] = sp1[r];      // o = m + 16
    }
    __syncthreads();  // RAW: spikes visible for A-operand reads

    // ---- layer 2: H2 = S_t(16x32) * W2^T(32x16pad) + b2  (8 WMMAs) ----
    v8f acc2;
#pragma unroll
    for (int r = 0; r < 8; ++r) acc2[r] = b2v;
    const float* scol = &sl[34 * half + m];
#pragma unroll
    for (int j = 0; j < 8; ++j) {
      v2f a; a.x = scol[68 * j]; a.y = scol[68 * j + 17];  // A2[m][4j+2h], A2[m][4j+1+2h]
      acc2 = __builtin_amdgcn_wmma_f32_16x16x4_f32(false, a, false, Bw2[j],
                                                   (short)0, acc2, false, false);
    }

    // ---- layer-2 IF neuron + firing-rate accumulation ----
#pragma unroll
    for (int r = 0; r < 8; ++r) {
      vm2[r] += acc2[r];
      if (vm2[r] >= 1.0f) { rate[r] += 1.0f; vm2[r] = 0.0f; }
    }
  }

  // ---- write mean firing rate: out[b][n], n = lane&15 < 10 ----
  if (m < NO) {
#pragma unroll
    for (int r = 0; r < 8; ++r) {
      out[(tb + (size_t)(r + 8 * half)) * NO + m] = rate[r] * (1.0f / 28.0f);
    }
  }
}

extern "C" void kernel_launch(void* const* d_in, const int* in_sizes, int n_in,
                              void* d_out, int out_size, void* d_ws, size_t ws_size,
                              hipStream_t stream) {
  const float* x  = (const float*)d_in[0];   // [B,1,28,28] fp32
  const float* W1 = (const float*)d_in[1];   // [32,28]
  const float* b1 = (const float*)d_in[2];   // [32]
  const float* W2 = (const float*)d_in[3];   // [10,32]
  const float* b2 = (const float*)d_in[4];   // [10]
  float* out = (float*)d_out;                // [B,10]

  const int B = in_sizes[0] / (NI * NT);     // 65536
  const int nblocks = B / BT;                // 4096, exact
  snn_if_mlp<<<nblocks, 32, 0, stream>>>(x, W1, b1, W2, b2, out);
}